// GraphEncoder_41755672051881
// MI455X (gfx1250) — compile-verified
//
#include <hip/hip_runtime.h>
#include <hip/hip_bf16.h>

#define NNODE 10000
#define NEDGE 320000
#define DIN 128
#define DMODEL 256
#define NHEAD 4
#define HDIM 64
#define NLAYER 2
#define DOUT 2048
#define KTOK 2
#define EMB 4096
#define BB 64
#define SS 512
#define BSZ (BB * SS)        /* 32768 flat positions */
#define KEMB (KTOK * EMB)    /* 8192 */

typedef __attribute__((ext_vector_type(2))) float v2f;
typedef __attribute__((ext_vector_type(4))) float v4f;
typedef __attribute__((ext_vector_type(8))) float v8f;

/* ---- CDNA5 async global->LDS copy helpers (ASYNCcnt-tracked) ----- */
__device__ __forceinline__ unsigned lds_off_u32(const void* p) {
  /* generic LDS address: addr[31:0] is the wave-relative LDS byte offset */
  return (unsigned)(unsigned long long)p;
}
__device__ __forceinline__ void async_copy_b128(unsigned lds_off, const float* gptr) {
  asm volatile("global_load_async_to_lds_b128 %0, %1, off"
               :: "v"(lds_off), "v"(gptr) : "memory");
}
__device__ __forceinline__ void wait_async0() {
  asm volatile("s_wait_asynccnt 0x0" ::: "memory");
}

/* ------------------------------------------------------------------ */
/* 1) exclusive-scan is_node -> node_pos[j] = flat index of j-th node */
/* ------------------------------------------------------------------ */
__global__ __launch_bounds__(1024) void scan_node_pos(const int* __restrict__ is_node,
                                                      int* __restrict__ node_pos,
                                                      int total) {
  __shared__ int sbuf[1024];
  __shared__ int srun;
  const int tid = threadIdx.x;
  if (tid == 0) srun = 0;
  __syncthreads();
  for (int base = 0; base < total; base += 1024) {
    int v = (base + tid < total) ? (is_node[base + tid] != 0) : 0;
    sbuf[tid] = v;
    __syncthreads();
    for (int off = 1; off < 1024; off <<= 1) {
      int t = (tid >= off) ? sbuf[tid - off] : 0;
      __syncthreads();
      sbuf[tid] += t;
      __syncthreads();
    }
    if (v) node_pos[srun + sbuf[tid] - 1] = base + tid;
    __syncthreads();
    if (tid == 0) srun += sbuf[1023];
    __syncthreads();
  }
}

/* ------------------------------------------------------------------ */
/* 2) non-node output rows: out[p,:] = embed_tokens[input_ids[p],:]   */
/*    streaming 537 MB -> non-temporal to keep L2 for reused tensors  */
/* ------------------------------------------------------------------ */
__global__ __launch_bounds__(256) void fill_base_embed(const int* __restrict__ input_ids,
                                                       const int* __restrict__ is_node,
                                                       const float* __restrict__ embtok,
                                                       float* __restrict__ out) {
  const int row = blockIdx.x;
  if (is_node[row]) return;  /* node rows written by fused GEMM scatter */
  const v4f* src = (const v4f*)(embtok + (size_t)input_ids[row] * EMB);
  v4f* dst = (v4f*)(out + (size_t)row * EMB);
  for (int i = threadIdx.x; i < EMB / 4; i += 256) {
    v4f v = __builtin_nontemporal_load(src + i);
    __builtin_nontemporal_store(v, dst + i);
  }
}

/* ------------------------------------------------------------------ */
/* fp32 WMMA GEMM: C[M,N] = A[M,K] @ B[K,N] (+bias[N])                */
/* block tile 64x64x16, 8 waves, wave tile 16x32 (two 16x16 accs).    */
/* Double-buffered LDS staged with global_load_async_to_lds_b128:     */
/* one barrier per tile, copy of tile k+1 overlaps WMMAs of tile k.   */
/* ------------------------------------------------------------------ */
#define BM 64
#define BN 64
#define BKT 16
#define LDA_S 20
#define LDB_S 68
#define A_TILE (BM * LDA_S)
#define B_TILE (BKT * LDB_S)

__device__ __forceinline__ void gemm_body(const float* __restrict__ A,
                                          const float* __restrict__ B,
                                          int M, int N, int K,
                                          float* As, float* Bs,
                                          v8f& acc0, v8f& acc1,
                                          int m0, int n0) {
  const int tid = threadIdx.x;
  const int lane = tid & 31;
  const int wave = tid >> 5;
  const int wm = wave & 3;
  const int wn = wave >> 2;
  const int half = lane >> 4;
  const int lm = lane & 15;

  /* cooperative staging: each thread owns one 16-byte chunk per tile */
  const int arow = tid >> 2;
  const int acol = (tid & 3) * 4;
  const int brow = tid >> 4;
  const int bcol = (tid & 15) * 4;
  const int aGRow = min(m0 + arow, M - 1);

  const unsigned ldsA = lds_off_u32(As + arow * LDA_S + acol);
  const unsigned ldsB = lds_off_u32(Bs + brow * LDB_S + bcol);
  const float* gA = A + (size_t)aGRow * K + acol;
  const float* gB = B + (size_t)brow * N + n0 + bcol;

  /* prologue: stage tile 0 into buffer 0 */
  async_copy_b128(ldsA, gA);
  async_copy_b128(ldsB, gB);

  int buf = 0;
  for (int k0 = 0; k0 < K; k0 += BKT) {
    wait_async0();      /* my async copies for this tile landed in LDS */
    __syncthreads();    /* everyone's landed; prev buffer fully read   */
    if (k0 + BKT < K) { /* kick next tile into the other buffer        */
      const int nb = buf ^ 1;
      async_copy_b128(ldsA + (unsigned)(nb * A_TILE * 4), gA + (k0 + BKT));
      async_copy_b128(ldsB + (unsigned)(nb * B_TILE * 4), gB + (size_t)(k0 + BKT) * N);
    }
    const float* Ab = As + buf * A_TILE;
    const float* Bb = Bs + buf * B_TILE;
#pragma unroll
    for (int kk = 0; kk < BKT; kk += 4) {
      const int ka = kk + 2 * half;             /* K-pair owned by this lane half */
      v2f a, b0, b1;
      a.x = Ab[(wm * 16 + lm) * LDA_S + ka];
      a.y = Ab[(wm * 16 + lm) * LDA_S + ka + 1];
      const int nb2 = wn * 32 + lm;
      b0.x = Bb[ka * LDB_S + nb2];
      b0.y = Bb[(ka + 1) * LDB_S + nb2];
      b1.x = Bb[ka * LDB_S + nb2 + 16];
      b1.y = Bb[(ka + 1) * LDB_S + nb2 + 16];
      acc0 = __builtin_amdgcn_wmma_f32_16x16x4_f32(false, a, false, b0, (short)0, acc0, false, false);
      acc1 = __builtin_amdgcn_wmma_f32_16x16x4_f32(false, a, false, b1, (short)0, acc1, false, false);
    }
    buf ^= 1;
  }
}

__global__ __launch_bounds__(256) void gemm_f32_wmma(const float* __restrict__ A,
                                                     const float* __restrict__ B,
                                                     const float* __restrict__ bias,
                                                     float* __restrict__ C,
                                                     int M, int N, int K) {
  __shared__ float As[2 * A_TILE];
  __shared__ float Bs[2 * B_TILE];
  const int m0 = blockIdx.y * BM;
  const int n0 = blockIdx.x * BN;
  v8f acc0 = {}, acc1 = {};
  gemm_body(A, B, M, N, K, As, Bs, acc0, acc1, m0, n0);

  const int lane = threadIdx.x & 31;
  const int wave = threadIdx.x >> 5;
  const int half = lane >> 4;
  const int lm = lane & 15;
  const int ccol = n0 + (wave >> 2) * 32 + lm;
  const float b0v = bias ? bias[ccol] : 0.f;
  const float b1v = bias ? bias[ccol + 16] : 0.f;
#pragma unroll
  for (int i = 0; i < 8; ++i) {
    const int row = m0 + (wave & 3) * 16 + half * 8 + i;
    if (row < M) {
      C[(size_t)row * N + ccol] = acc0[i] + b0v;
      C[(size_t)row * N + ccol + 16] = acc1[i] + b1v;
    }
  }
}

/* Same GEMM, epilogue scatters node rows straight into d_out.
   Output column c of node n maps to out[node_pos[2n + c/4096], c%4096]. */
__global__ __launch_bounds__(256) void gemm_f32_wmma_scatter(const float* __restrict__ A,
                                                             const float* __restrict__ B,
                                                             const float* __restrict__ bias,
                                                             const int* __restrict__ node_pos,
                                                             float* __restrict__ out,
                                                             int M, int N, int K) {
  __shared__ float As[2 * A_TILE];
  __shared__ float Bs[2 * B_TILE];
  const int m0 = blockIdx.y * BM;
  const int n0 = blockIdx.x * BN;
  v8f acc0 = {}, acc1 = {};
  gemm_body(A, B, M, N, K, As, Bs, acc0, acc1, m0, n0);

  const int lane = threadIdx.x & 31;
  const int wave = threadIdx.x >> 5;
  const int half = lane >> 4;
  const int lm = lane & 15;
  const int c0 = n0 + (wave >> 2) * 32 + lm;
  const int c1 = c0 + 16;
  const float b0v = bias[c0];
  const float b1v = bias[c1];
#pragma unroll
  for (int i = 0; i < 8; ++i) {
    const int row = m0 + (wave & 3) * 16 + half * 8 + i;
    if (row < M) {
      const int r0 = node_pos[row * KTOK + (c0 >> 12)];
      const int r1 = node_pos[row * KTOK + (c1 >> 12)];
      __builtin_nontemporal_store(acc0[i] + b0v, out + (size_t)r0 * EMB + (c0 & (EMB - 1)));
      __builtin_nontemporal_store(acc1[i] + b1v, out + (size_t)r1 * EMB + (c1 & (EMB - 1)));
    }
  }
}

/* ------------------------------------------------------------------ */
/* graph attention kernels (all node tensors are L2-resident)         */
/* ------------------------------------------------------------------ */
__global__ __launch_bounds__(256) void zero_f32(float* __restrict__ p, int n) {
  int i = blockIdx.x * 256 + threadIdx.x;
  if (i < n) p[i] = 0.f;
}

__global__ __launch_bounds__(256) void edge_attention(const float* __restrict__ qn,
                                                      const float* __restrict__ kn,
                                                      const int* __restrict__ rows,
                                                      const int* __restrict__ cols,
                                                      float* __restrict__ exp_att,
                                                      float* __restrict__ att_norm) {
  int g = blockIdx.x * 256 + threadIdx.x;
  if (g >= NEDGE * NHEAD) return;
  const int e = g >> 2;
  const int h = g & 3;
  const int r = rows[e], c = cols[e];
  const float4* q4 = (const float4*)(qn + (size_t)r * DMODEL + h * HDIM);
  const float4* k4 = (const float4*)(kn + (size_t)c * DMODEL + h * HDIM);
  float dot = 0.f;
#pragma unroll
  for (int i = 0; i < HDIM / 4; ++i) {
    float4 q = q4[i], k = k4[i];
    dot += q.x * k.x + q.y * k.y + q.z * k.z + q.w * k.w;
  }
  dot = fminf(10.f, fmaxf(-10.f, dot));
  float ea = expf(dot);
  exp_att[g] = ea;
  atomicAdd(att_norm + r * NHEAD + h, ea);
}

__global__ __launch_bounds__(256) void edge_aggregate(const float* __restrict__ vn,
                                                      const int* __restrict__ rows,
                                                      const int* __restrict__ cols,
                                                      const float* __restrict__ exp_att,
                                                      const float* __restrict__ att_norm,
                                                      float* __restrict__ res) {
  int g = blockIdx.x * 256 + threadIdx.x;
  if (g >= NEDGE * (DMODEL / 4)) return;
  const int e = g >> 6;
  const int c4 = g & 63;
  const int h = c4 >> 4;
  const int r = rows[e];
  const float a = exp_att[e * NHEAD + h] / (att_norm[r * NHEAD + h] + 1e-8f);
  const float4 vv = *(const float4*)(vn + (size_t)cols[e] * DMODEL + c4 * 4);
  float* dst = res + (size_t)r * DMODEL + c4 * 4;
  atomicAdd(dst + 0, a * vv.x);
  atomicAdd(dst + 1, a * vv.y);
  atomicAdd(dst + 2, a * vv.z);
  atomicAdd(dst + 3, a * vv.w);
}

__global__ __launch_bounds__(256) void residual_layernorm(const float* __restrict__ res,
                                                          const float* __restrict__ embIn,
                                                          const float* __restrict__ w,
                                                          const float* __restrict__ b,
                                                          float* __restrict__ embOut) {
  __shared__ float sred[256];
  const int row = blockIdx.x;
  const int tid = threadIdx.x;
  const float v = res[(size_t)row * DMODEL + tid] + embIn[(size_t)row * DMODEL + tid];
  sred[tid] = v;
  __syncthreads();
  for (int off = 128; off > 0; off >>= 1) {
    if (tid < off) sred[tid] += sred[tid + off];
    __syncthreads();
  }
  const float mu = sred[0] * (1.0f / DMODEL);
  __syncthreads();
  const float d = v - mu;
  sred[tid] = d * d;
  __syncthreads();
  for (int off = 128; off > 0; off >>= 1) {
    if (tid < off) sred[tid] += sred[tid + off];
    __syncthreads();
  }
  const float var = sred[0] * (1.0f / DMODEL);
  embOut[(size_t)row * DMODEL + tid] = d * rsqrtf(var + 1e-6f) * w[tid] + b[tid];
}

/* bias_c = inv_b @ proj_w + proj_b */
__global__ __launch_bounds__(256) void fold_bias(const float* __restrict__ inv_b,
                                                 const float* __restrict__ proj_w,
                                                 const float* __restrict__ proj_b,
                                                 float* __restrict__ bias_c) {
  const int j = blockIdx.x * 256 + threadIdx.x;
  if (j >= KEMB) return;
  float s = proj_b[j];
  for (int t = 0; t < DOUT; ++t) s += inv_b[t] * proj_w[(size_t)t * KEMB + j];
  bias_c[j] = s;
}

/* ------------------------------------------------------------------ */
extern "C" void kernel_launch(void* const* d_in, const int* in_sizes, int n_in,
                              void* d_out, int out_size, void* d_ws, size_t ws_size,
                              hipStream_t stream) {
  const int*   input_ids = (const int*)d_in[0];
  const int*   is_node   = (const int*)d_in[1];
  const float* x         = (const float*)d_in[2];
  const int*   edge      = (const int*)d_in[3];
  const float* W_P_w     = (const float*)d_in[4];
  const float* W_P_b     = (const float*)d_in[5];
  const float* Wq        = (const float*)d_in[6];
  const float* Wk        = (const float*)d_in[7];
  const float* Wv        = (const float*)d_in[8];
  const float* ln_w      = (const float*)d_in[9];
  const float* ln_b      = (const float*)d_in[10];
  const float* inv_w     = (const float*)d_in[11];
  const float* inv_b     = (const float*)d_in[12];
  const float* proj_w    = (const float*)d_in[13];
  const float* proj_b    = (const float*)d_in[14];
  const float* embtok    = (const float*)d_in[15];
  float* out = (float*)d_out;

  const int* rows = edge;
  const int* cols = edge + NEDGE;

  /* workspace layout (floats) */
  const size_t ND = (size_t)NNODE * DMODEL;  /* 2.56 M floats */
  float* ws    = (float*)d_ws;
  float* embA  = ws;
  float* embB  = embA + ND;
  float* qn    = embB + ND;
  float* kn    = qn + ND;
  float* vn    = kn + ND;
  float* res   = vn + ND;
  float* expat = res + ND;                              /* E*H   */
  float* norm  = expat + (size_t)NEDGE * NHEAD;         /* N*H   */
  float* biasc = norm + (size_t)NNODE * NHEAD;          /* KEMB  */
  int*   npos  = (int*)(biasc + KEMB);                  /* N*K   */
  float* Wc    = qn; /* reuse q/k/v region after the graph layers */

  const int mTiles = (NNODE + BM - 1) / BM;             /* 157 */

  /* 1) node-token positions */
  scan_node_pos<<<1, 1024, 0, stream>>>(is_node, npos, BSZ);
  /* 2) non-node output rows (537 MB stream, the HBM floor) */
  fill_base_embed<<<BSZ, 256, 0, stream>>>(input_ids, is_node, embtok, out);
  /* 3) input projection: embA = x @ W_P_w + b */
  gemm_f32_wmma<<<dim3(DMODEL / BN, mTiles), 256, 0, stream>>>(
      x, W_P_w, W_P_b, embA, NNODE, DMODEL, DIN);

  /* 4) graph transformer layers */
  for (int l = 0; l < NLAYER; ++l) {
    const float* eI = (l == 0) ? embA : embB;
    float*       eO = (l == 0) ? embB : embA;
    const size_t wOff = (size_t)l * DMODEL * DMODEL;
    gemm_f32_wmma<<<dim3(DMODEL / BN, mTiles), 256, 0, stream>>>(
        eI, Wq + wOff, nullptr, qn, NNODE, DMODEL, DMODEL);
    gemm_f32_wmma<<<dim3(DMODEL / BN, mTiles), 256, 0, stream>>>(
        eI, Wk + wOff, nullptr, kn, NNODE, DMODEL, DMODEL);
    gemm_f32_wmma<<<dim3(DMODEL / BN, mTiles), 256, 0, stream>>>(
        eI, Wv + wOff, nullptr, vn, NNODE, DMODEL, DMODEL);
    zero_f32<<<(int)((ND + 255) / 256), 256, 0, stream>>>(res, (int)ND);
    zero_f32<<<(NNODE * NHEAD + 255) / 256, 256, 0, stream>>>(norm, NNODE * NHEAD);
    edge_attention<<<(NEDGE * NHEAD + 255) / 256, 256, 0, stream>>>(
        qn, kn, rows, cols, expat, norm);
    edge_aggregate<<<(NEDGE * (DMODEL / 4) + 255) / 256, 256, 0, stream>>>(
        vn, rows, cols, expat, norm, res);
    residual_layernorm<<<NNODE, 256, 0, stream>>>(
        res, eI, ln_w + l * DMODEL, ln_b + l * DMODEL, eO);
  }

  /* 5) Wc = inv_w @ proj_w  (algebraic fusion: 346 -> 51 GFLOP total) */
  gemm_f32_wmma<<<dim3(KEMB / BN, DMODEL / BM), 256, 0, stream>>>(
      inv_w, proj_w, nullptr, Wc, DMODEL, KEMB, DOUT);
  /* 6) bias_c = inv_b @ proj_w + proj_b */
  fold_bias<<<KEMB / 256, 256, 0, stream>>>(inv_b, proj_w, proj_b, biasc);
  /* 7) node rows: out[node_pos] = embA @ Wc + bias_c (fused scatter) */
  gemm_f32_wmma_scatter<<<dim3(KEMB / BN, mTiles), 256, 0, stream>>>(
      embA, Wc, biasc, npos, out, NNODE, KEMB, DMODEL);
}